// ZCANormSVDPI_37151467110665
// MI455X (gfx1250) — compile-verified
//
#include <hip/hip_runtime.h>

typedef __attribute__((ext_vector_type(2))) float v2f;
typedef __attribute__((ext_vector_type(4))) float v4f;
typedef __attribute__((ext_vector_type(8))) float v8f;

#define B_SZ 1024
#define C_SZ 64
#define M_SZ 512
#define TM 4            // clusters (m) per block
#define TB 64           // batch rows per block
#define DPAD 68         // padded row pitch (68 % 64 == 4 -> conflict-free b64 frag reads)
#define MATSZ (C_SZ * DPAD)   // 4352 floats per staged 64x64 matrix slab

__global__ __launch_bounds__(256) void zca_wmma_kernel(
    const float* __restrict__ x,      // (B, C, M)
    const float* __restrict__ rm,     // (1, C, M)
    const float* __restrict__ S,      // (M, C, C)
    float* __restrict__ out)          // (B, C, M)
{
  extern __shared__ float lds[];
  float* ss = lds;                 // [TM][C][DPAD] : S[m] row-major (c, d)
  float* xt = lds + TM * MATSZ;    // [TM][TB][DPAD]: xc tile (b, d)

  const int t  = threadIdx.x;
  const int mt = blockIdx.x & 127;     // 128 m-tiles
  const int bt = blockIdx.x >> 7;      // 16 b-tiles
  const int m0 = mt * TM;
  const int b0 = bt * TB;

  // ---- stage S[m0..m0+3] : 4 * 4096 floats, one float4 per thread per pass (coalesced)
  #pragma unroll
  for (int p = 0; p < 16; ++p) {
    const int j     = p & 3;
    const int chunk = p >> 2;
    const int off   = chunk * 1024 + t * 4;        // 0..4095, 16B aligned
    v4f s4 = *(const v4f*)(S + (m0 + j) * (C_SZ * C_SZ) + off);
    const int c = off >> 6;        // row (c)
    const int d = off & 63;        // col (d), multiple of 4
    *(v4f*)(ss + j * MATSZ + c * DPAD + d) = s4;
  }

  // ---- stage xc tile: b in [b0,b0+64), d in [0,64), m in [m0,m0+4)
  //      float4 over the 4 consecutive m's (global_load_b128), transpose into LDS [m][b][d]
  #pragma unroll
  for (int p = 0; p < 16; ++p) {
    const int b = p * 4 + (t >> 6);
    const int d = t & 63;
    v4f x4 = *(const v4f*)(x  + (b0 + b) * (C_SZ * M_SZ) + d * M_SZ + m0);
    v4f r4 = *(const v4f*)(rm +                            d * M_SZ + m0);
    v4f xc = x4 - r4;
    const int bi = b * DPAD + d;
    xt[0 * MATSZ + bi] = xc.x;
    xt[1 * MATSZ + bi] = xc.y;
    xt[2 * MATSZ + bi] = xc.z;
    xt[3 * MATSZ + bi] = xc.w;
  }

  __syncthreads();

  const int w  = t >> 5;     // wave id (0..7)
  const int l  = t & 31;     // lane
  const int lo = l & 15;
  const int hi = l >> 4;

  // 64 output tiles (4 m  x  4 b-subtiles  x  4 c-subtiles); 8 tiles per wave
  for (int i = 0; i < 8; ++i) {
    const int tile = w + i * 8;
    const int mj   = tile & 3;
    const int bt2  = (tile >> 2) & 3;
    const int ct   = tile >> 4;

    // A (xc): lanes 0-15 = rows b, VGPR0/1 hold K = 2*hi, 2*hi+1
    const float* Ab = xt + mj * MATSZ + (bt2 * 16 + lo) * DPAD + 2 * hi;
    // B (S^T): lanes 0-15 = cols c; element (K=d, N=c) = S[m][c][d]
    const float* Bb = ss + mj * MATSZ + (ct * 16 + lo) * DPAD + 2 * hi;

    v8f acc = {};
    #pragma unroll
    for (int k = 0; k < 64; k += 4) {
      v2f a = *(const v2f*)(Ab + k);   // ds_load_b64, conflict-free
      v2f b = *(const v2f*)(Bb + k);
      acc = __builtin_amdgcn_wmma_f32_16x16x4_f32(
          /*neg_a=*/false, a, /*neg_b=*/false, b,
          /*c_mod=*/(short)0, acc, /*reuse_a=*/false, /*reuse_b=*/false);
    }

    // D layout: VGPR r -> row (b) = r + 8*hi, col (c) = lo
    const int c_out  = ct * 16 + lo;
    const int b_base = b0 + bt2 * 16 + 8 * hi;
    float* op = out + c_out * M_SZ + (m0 + mj);
    #pragma unroll
    for (int r = 0; r < 8; ++r)
      op[(b_base + r) * (C_SZ * M_SZ)] = acc[r];
  }
}

extern "C" void kernel_launch(void* const* d_in, const int* in_sizes, int n_in,
                              void* d_out, int out_size, void* d_ws, size_t ws_size,
                              hipStream_t stream) {
  (void)in_sizes; (void)n_in; (void)d_ws; (void)ws_size; (void)out_size;
  const float* x  = (const float*)d_in[0];   // (1024, 64, 512)
  const float* rm = (const float*)d_in[1];   // (1, 64, 512)
  const float* S  = (const float*)d_in[2];   // (512, 64, 64)
  float* out = (float*)d_out;                // (1024, 64, 512)

  const int grid = (M_SZ / TM) * (B_SZ / TB);          // 128 * 16 = 2048 blocks
  const size_t smem = (size_t)2 * TM * MATSZ * sizeof(float);  // 136 KB dynamic LDS
  zca_wmma_kernel<<<grid, 256, smem, stream>>>(x, rm, S, out);
}